// Network_29076928594528
// MI455X (gfx1250) — compile-verified
//
#include <hip/hip_runtime.h>

// Triplane sample + MLP(32->128->128->4, ReLU) fused, fp32 WMMA (16x16x4).
// B=4, N=262144 -> 1,048,576 points = 65,536 wave-tiles of 16 points.
// Round 2: bound unrolling of the output-tile loops to cap VGPR pressure
// (round 1 allocated past v255 -> s_set_vgpr_msb, killing occupancy).

#define HW      (128 * 128)
#define PLANE   (32 * 128 * 128)
#define WAVES   8

typedef __attribute__((ext_vector_type(2))) float v2f;
typedef __attribute__((ext_vector_type(8))) float v8f;

__device__ __forceinline__ v8f wmma4(v2f a, v2f b, v8f c) {
    // D = A(16x4 f32) * B(4x16 f32) + C(16x16 f32), exact fp32 MAC on matrix pipe
    return __builtin_amdgcn_wmma_f32_16x16x4_f32(
        /*neg_a=*/false, a, /*neg_b=*/false, b,
        /*c_mod=*/(short)0, c, /*reuse_a=*/false, /*reuse_b=*/false);
}

// Bilinear sample (align_corners=True, zeros padding) of 16 channels of one
// 32x128x128 plane, accumulated into the f32 WMMA A-fragment register layout:
// fa[s][j] holds channel 4*s + 2*half + j for point m = lane&15.
__device__ __forceinline__ void sample_plane(const float* __restrict__ pb,
                                             float gx, float gy, int half,
                                             float (&fa)[8][2]) {
    float fx = (gx + 1.0f) * 0.5f * 127.0f;
    float fy = (gy + 1.0f) * 0.5f * 127.0f;
    int ix = (int)floorf(fx); ix = ix < 0 ? 0 : (ix > 127 ? 127 : ix);
    int iy = (int)floorf(fy); iy = iy < 0 ? 0 : (iy > 127 ? 127 : iy);
    float wx = fx - (float)ix;          // == 0 when ix hits the right edge
    float wy = fy - (float)iy;          // == 0 when iy hits the bottom edge
    bool sx = ix > 126;                 // at edge: shift load window left by 1
    int  xs = sx ? 126 : ix;
    int  y1 = iy > 126 ? 127 : iy + 1;  // clamped; weight is 0 there anyway
    int  o0 = iy * 128 + xs;
    int  o1 = y1 * 128 + xs;
    float w00 = (1.f - wx) * (1.f - wy);
    float w01 = wx * (1.f - wy);
    float w10 = (1.f - wx) * wy;
    float w11 = wx * wy;
#pragma unroll
    for (int s = 0; s < 8; ++s) {
#pragma unroll
        for (int j = 0; j < 2; ++j) {
            const float* p = pb + (4 * s + 2 * half + j) * HW;
            float a0 = p[o0],     a1 = p[o0 + 1];
            float b0 = p[o1],     b1 = p[o1 + 1];
            float c00 = sx ? a1 : a0;
            float c10 = sx ? b1 : b0;
            fa[s][j] += w00 * c00 + w01 * a1 + w10 * c10 + w11 * b1;
        }
    }
}

__global__ void __launch_bounds__(WAVES * 32)
triplane_mlp_kernel(const float* __restrict__ gc,   // (B,N,3)
                    const float* __restrict__ tp,   // (96,128,128): xy | yz | xz
                    const float* __restrict__ w0, const float* __restrict__ b0,
                    const float* __restrict__ w1, const float* __restrict__ b1,
                    const float* __restrict__ w2, const float* __restrict__ b2,
                    float* __restrict__ out)        // (B,N,4)
{
    __shared__ float Hsh[WAVES][16 * 128];          // 64 KB: per-wave activation tile

    const int tid  = threadIdx.x;
    const int wid  = tid >> 5;
    const int lane = tid & 31;
    const int half = lane >> 4;     // which 16-lane half of the wave
    const int m    = lane & 15;     // point row (A/M) and output col (B/N) index
    float* Hs = Hsh[wid];

    const int tile = blockIdx.x * WAVES + wid;
    const int base = tile * 16;                      // first point of this wave
    const int g    = base + m;

    const float x = gc[3 * g + 0];
    const float y = gc[3 * g + 1];
    const float z = gc[3 * g + 2];

    // ---- Phase 1: triplane gather straight into layer-0 A fragments --------
    float fa[8][2];
#pragma unroll
    for (int s = 0; s < 8; ++s) { fa[s][0] = 0.f; fa[s][1] = 0.f; }
    sample_plane(tp + 0 * PLANE, x, y, half, fa);    // feat_xy(x, y)
    sample_plane(tp + 2 * PLANE, x, z, half, fa);    // feat_xz(x, z)
    sample_plane(tp + 1 * PLANE, y, z, half, fa);    // feat_yz(y, z)

    // ---- Layer 0: (16x32) @ (32x128) + b0, ReLU -> LDS ---------------------
    // unroll 2: two independent accumulator chains, bounded register pressure
#pragma unroll 2
    for (int t = 0; t < 8; ++t) {
        const int n0 = 16 * t;
        v8f acc = {0.f, 0.f, 0.f, 0.f, 0.f, 0.f, 0.f, 0.f};
#pragma unroll
        for (int s = 0; s < 8; ++s) {
            v2f a; a.x = fa[s][0]; a.y = fa[s][1];
            v2f b = *(const v2f*)(w0 + (n0 + m) * 32 + 4 * s + 2 * half);
            acc = wmma4(a, b, acc);
        }
        const float bias = b0[n0 + m];
#pragma unroll
        for (int r = 0; r < 8; ++r) {
            float h = acc[r] + bias;
            h = h > 0.f ? h : 0.f;
            Hs[(r + 8 * half) * 128 + n0 + m] = h;   // C/D layout -> [M][N]
        }
    }
    __syncthreads();

    // ---- Gather layer-1 A fragments (16x128) into registers ----------------
    float ha[32][2];
#pragma unroll
    for (int s = 0; s < 32; ++s) {
        v2f v = *(const v2f*)(Hs + m * 128 + 4 * s + 2 * half);  // ds_load_b64
        ha[s][0] = v.x; ha[s][1] = v.y;
    }
    __syncthreads();

    // ---- Layer 1: (16x128) @ (128x128) + b1, ReLU -> LDS (buffer reused) ---
#pragma unroll 2
    for (int t = 0; t < 8; ++t) {
        const int n0 = 16 * t;
        v8f acc = {0.f, 0.f, 0.f, 0.f, 0.f, 0.f, 0.f, 0.f};
#pragma unroll
        for (int s = 0; s < 32; ++s) {
            v2f a; a.x = ha[s][0]; a.y = ha[s][1];
            v2f b = *(const v2f*)(w1 + (n0 + m) * 128 + 4 * s + 2 * half);
            acc = wmma4(a, b, acc);
        }
        const float bias = b1[n0 + m];
#pragma unroll
        for (int r = 0; r < 8; ++r) {
            float h = acc[r] + bias;
            h = h > 0.f ? h : 0.f;
            Hs[(r + 8 * half) * 128 + n0 + m] = h;
        }
    }
    __syncthreads();

    // ---- Layer 2: (16x128) @ (128x4) in one 16x16 tile (cols 0..3 valid) ---
#pragma unroll
    for (int s = 0; s < 32; ++s) {
        v2f v = *(const v2f*)(Hs + m * 128 + 4 * s + 2 * half);
        ha[s][0] = v.x; ha[s][1] = v.y;
    }
    const int wrow = m < 3 ? m : 3;                  // clamp for in-bounds loads
    v8f acc = {0.f, 0.f, 0.f, 0.f, 0.f, 0.f, 0.f, 0.f};
#pragma unroll
    for (int s = 0; s < 32; ++s) {
        v2f a; a.x = ha[s][0]; a.y = ha[s][1];
        v2f b = *(const v2f*)(w2 + wrow * 128 + 4 * s + 2 * half);
        acc = wmma4(a, b, acc);
    }

    // ---- Bias + ReLU + channel permute [1,2,3,0] and store -----------------
    // For fixed r, lanes m=0..3 of each half write 4 consecutive floats of one
    // point (permuted within the 16B line) -> hardware-coalesced.
    if (m < 4) {                                     // predication only after WMMAs
        const float bias = b2[m];
        const int   oc   = (m + 3) & 3;              // net ch m -> out ch (m+3)%4
#pragma unroll
        for (int r = 0; r < 8; ++r) {
            float v = acc[r] + bias;
            v = v > 0.f ? v : 0.f;
            const int pt = base + r + 8 * half;
            out[pt * 4 + oc] = v;
        }
    }
}

extern "C" void kernel_launch(void* const* d_in, const int* in_sizes, int n_in,
                              void* d_out, int out_size, void* d_ws, size_t ws_size,
                              hipStream_t stream) {
    const float* gc = (const float*)d_in[0];   // (4, 262144, 3)
    const float* tp = (const float*)d_in[1];   // (96, 128, 128)
    const float* w0 = (const float*)d_in[2];
    const float* b0 = (const float*)d_in[3];
    const float* w1 = (const float*)d_in[4];
    const float* b1 = (const float*)d_in[5];
    const float* w2 = (const float*)d_in[6];
    const float* b2 = (const float*)d_in[7];
    float* out = (float*)d_out;

    const int n_pts  = in_sizes[0] / 3;        // 1,048,576
    const int tiles  = n_pts / 16;             // 65,536 (divides exactly)
    const int blocks = tiles / WAVES;          // 8,192

    triplane_mlp_kernel<<<blocks, WAVES * 32, 0, stream>>>(
        gc, tp, w0, b0, w1, b1, w2, b2, out);
}